// DualHeadDecoder_68685116997965
// MI455X (gfx1250) — compile-verified
//
#include <hip/hip_runtime.h>
#include <hip/hip_bf16.h>

typedef float v2f __attribute__((ext_vector_type(2)));
typedef float v8f __attribute__((ext_vector_type(8)));

#define WAVES_PER_BLOCK 2
#define TILE_E 16      // edges per wave tile (M of the WMMA)
#define DD 128         // feature dim D
#define D2 256         // 2*D  (K of the main GEMM)
#define HH 16          // hidden dim (N of the WMMA)

// lane-xor exchange via ds_swizzle (group-of-32: offset = xor<<10 | and_mask 0x1F)
#define SWZ_XOR(x, m) \
    __int_as_float(__builtin_amdgcn_ds_swizzle(__float_as_int(x), (((m) << 10) | 0x1F)))

__global__ __launch_bounds__(64) void dual_head_decoder_kernel(
    const float* __restrict__ z,           // [N,128]
    const int*   __restrict__ edge_index,  // [2,E] flattened: rows then cols
    const float* __restrict__ W1,          // [256,16] row-major ([in,out])
    const float* __restrict__ b1,          // [16]
    const float* __restrict__ W2,          // [16]
    const float* __restrict__ b2,          // [1]
    float* __restrict__ out_logits,        // [E]
    float* __restrict__ out_w,             // [E]
    int E, int numTiles)
{
    __shared__ float sW1t[HH * D2];                     // 16 KB: W1 transposed, [n][k]
    __shared__ float sA[WAVES_PER_BLOCK][TILE_E * D2];  // 2 x 16 KB: per-wave A tiles

    const int tid  = threadIdx.x;
    const int wave = tid >> 5;
    const int lane = tid & 31;

    // ---- cooperative transposed load of W1 into LDS (reused by all tiles)
    for (int i = tid; i < HH * D2; i += blockDim.x) {
        const int k = i >> 4;   // 0..255
        const int n = i & 15;   // 0..15
        sW1t[n * D2 + k] = W1[i];
    }

    const int tile  = blockIdx.x * WAVES_PER_BLOCK + wave;
    const int ebase = tile * TILE_E;
    float* aw = sA[wave];

    // ---- Phase 1: pure staging — gather 16 edges' z_row/z_col into LDS
    if (tile < numTiles) {
        // lanes 0..15 hold edge e = ebase+lane; lanes 16..31 mirror (clamped, no divergence)
        const int eIdx  = ebase + (lane & 15);
        const int ce    = (eIdx < E) ? eIdx : (E - 1);
        const int myRow = edge_index[ce];
        const int myCol = edge_index[E + ce];

        #pragma unroll 8
        for (int e = 0; e < TILE_E; ++e) {
            const int r = __shfl(myRow, e, 32);   // v_readlane (uniform)
            const int c = __shfl(myCol, e, 32);
            // full-wave coalesced 128-float row loads (32 lanes x float4)
            const float4 zr = *(const float4*)(z + (size_t)r * DD + lane * 4);
            const float4 zc = *(const float4*)(z + (size_t)c * DD + lane * 4);
            *(float4*)(aw + e * D2 +      lane * 4) = zr;   // z_row -> A[:,0:128]
            *(float4*)(aw + e * D2 + DD + lane * 4) = zc;   // z_col -> A[:,128:256]
        }
    }

    __syncthreads();  // sW1t visible to all waves (reached uniformly by the block)

    // ---- Phase 2: fused WMMA — main GEMM + dot-product head
    //   acc  += A_pair[16x256] @ W1[256x16]          (h pre-activation)
    //   dotC += Z_row[16x128]  @ Z_col^T[128x16]     (diag = adj_logits)
    // The B-fragment of the dot GEMM is bit-identical to the A-fragment of the
    // z_col half (lane index serves as M for A-frags and N for B-frags).
    if (tile < numTiles) {
        const int mn  = lane & 15;            // M for A-frag, N for B-frag
        const int sel = (lane >> 4) << 1;     // lanes 0-15 -> K+0/K+1, lanes 16-31 -> K+2/K+3
        const float* aBase = aw   + mn * D2 + sel;
        const float* bBase = sW1t + mn * D2 + sel;

        v8f acc  = {0.f, 0.f, 0.f, 0.f, 0.f, 0.f, 0.f, 0.f};
        v8f dotC = {0.f, 0.f, 0.f, 0.f, 0.f, 0.f, 0.f, 0.f};

        #pragma unroll 4
        for (int kc = 0; kc < DD / 4; ++kc) {   // 32 iters, 3 WMMAs each
            const v2f aR = *(const v2f*)(aBase + kc * 4);          // z_row K-chunk
            const v2f aC = *(const v2f*)(aBase + DD + kc * 4);     // z_col K-chunk
            const v2f bR = *(const v2f*)(bBase + kc * 4);          // W1 rows [0,128)
            const v2f bC = *(const v2f*)(bBase + DD + kc * 4);     // W1 rows [128,256)
            acc  = __builtin_amdgcn_wmma_f32_16x16x4_f32(false, aR, false, bR,
                                                         (short)0, acc,  false, false);
            acc  = __builtin_amdgcn_wmma_f32_16x16x4_f32(false, aC, false, bC,
                                                         (short)0, acc,  false, false);
            dotC = __builtin_amdgcn_wmma_f32_16x16x4_f32(false, aR, false, aC,
                                                         (short)0, dotC, false, false);
        }

        // ---- adj_logits: extract diagonal of dotC
        // C/D layout: VGPR j, lane l -> M = j + 8*(l>=16), N = l&15
        // diag(e): e<8 -> VGPR e, lane e ; e>=8 -> VGPR e-8, lane e+16
        {
            const int eLane = (lane < 8) ? lane : ((lane >= 24) ? (lane - 16) : -1);
            const int j0 = eLane & 7;
            float dv = dotC[0];
            #pragma unroll
            for (int j = 1; j < 8; ++j) dv = (j0 == j) ? dotC[j] : dv;
            if (eLane >= 0 && (ebase + eLane) < E) out_logits[ebase + eLane] = dv;
        }

        // ---- weights head: bias + ReLU, h @ W2 + b2, softplus
        const float b1v = b1[mn];
        const float w2v = W2[mn];
        const float b2v = b2[0];
        #pragma unroll
        for (int j = 0; j < 8; ++j) {
            float h = acc[j] + b1v;
            h = fmaxf(h, 0.0f);
            float t = h * w2v;
            // 16-lane half-wave sum via xor swizzles (stays within each half)
            t += SWZ_XOR(t, 1);
            t += SWZ_XOR(t, 2);
            t += SWZ_XOR(t, 4);
            t += SWZ_XOR(t, 8);
            const int eM = ebase + j + ((lane >> 4) << 3);
            if (mn == 0 && eM < E) {
                const float x = t + b2v;
                // numerically stable softplus
                out_w[eM] = fmaxf(x, 0.0f) + log1pf(expf(-fabsf(x)));
            }
        }
    }
}

extern "C" void kernel_launch(void* const* d_in, const int* in_sizes, int n_in,
                              void* d_out, int out_size, void* d_ws, size_t ws_size,
                              hipStream_t stream) {
    const float* z   = (const float*)d_in[0];
    const int*   ei  = (const int*)  d_in[1];
    const float* W1  = (const float*)d_in[2];
    const float* b1  = (const float*)d_in[3];
    const float* W2  = (const float*)d_in[4];
    const float* b2  = (const float*)d_in[5];

    const int E = in_sizes[1] / 2;          // edge_index is [2,E]
    float* out_logits = (float*)d_out;      // outputs concatenated: [E] logits, [E] weights
    float* out_w      = out_logits + E;

    const int numTiles = (E + TILE_E - 1) / TILE_E;
    const int blocks   = (numTiles + WAVES_PER_BLOCK - 1) / WAVES_PER_BLOCK;

    dual_head_decoder_kernel<<<blocks, 32 * WAVES_PER_BLOCK, 0, stream>>>(
        z, ei, W1, b1, W2, b2, out_logits, out_w, E, numTiles);
}